// CustomQuaternionLoss_1675037245668
// MI455X (gfx1250) — compile-verified
//
#include <hip/hip_runtime.h>
#include <hip/hip_bf16.h>

// CDNA5 / gfx1250 quaternion Huber loss reduction.
// Kernel 1: bandwidth-bound streaming pass (float4 loads -> global_load_b128),
//           per-block deterministic reduction into d_ws partials.
// Kernel 2: single wave32; WMMA f32 16x16x4 with A=ones performs a 64-way sum
//           of the per-lane accumulators (layout-agnostic since A is all-ones),
//           finished with 4 shfl_xor; writes final mean.

typedef __attribute__((ext_vector_type(2))) float v2f;
typedef __attribute__((ext_vector_type(8))) float v8f;

#define EPS_ANGLE 0.001f

__device__ __forceinline__ float huber1(float x) {
    float ax = fabsf(x);
    return (ax <= 1.0f) ? (0.5f * x * x) : (ax - 0.5f);
}

__global__ void quat_loss_partial(const float4* __restrict__ tq,
                                  const float4* __restrict__ pq,
                                  float* __restrict__ partial,
                                  int n) {
    int tid    = blockIdx.x * blockDim.x + threadIdx.x;
    int stride = gridDim.x * blockDim.x;

    float acc = 0.0f;
    for (int i = tid; i < n; i += stride) {
        float4 t = tq[i];   // 16B -> global_load_b128
        float4 p = pq[i];   // 16B -> global_load_b128

        // normalize both quaternions
        float tin = rsqrtf(t.x * t.x + t.y * t.y + t.z * t.z + t.w * t.w);
        float pin = rsqrtf(p.x * p.x + p.y * p.y + p.z * p.z + p.w * p.w);
        float tx = t.x * tin, ty = t.y * tin, tz = t.z * tin, tw = t.w * tin;
        float px = p.x * pin, py = p.y * pin, pz = p.z * pin, pw = p.w * pin;

        // d = p * conj(t); conj(t) = (-tx,-ty,-tz, tw)
        float qx = -tx, qy = -ty, qz = -tz, qw = tw;
        float rx = pw * qx + qw * px + (py * qz - pz * qy);
        float ry = pw * qy + qw * py + (pz * qx - px * qz);
        float rz = pw * qz + qw * pz + (px * qy - py * qx);
        float rw = pw * qw - (px * qx + py * qy + pz * qz);

        // rotvec: flip to rw >= 0
        float s  = (rw < 0.0f) ? -1.0f : 1.0f;
        rx *= s; ry *= s; rz *= s; rw *= s;

        float vn    = sqrtf(rx * rx + ry * ry + rz * rz);
        float ang   = 2.0f * atan2f(vn, rw);
        float scale;
        if (ang < EPS_ANGLE) {
            scale = 2.0f + ang * ang * (1.0f / 12.0f);
        } else {
            scale = ang / __sinf(0.5f * ang);
        }

        float vx = scale * rx, vy = scale * ry, vz = scale * rz;
        acc += huber1(vx) + huber1(vy) + huber1(vz);
    }

    // wave32 reduction
    #pragma unroll
    for (int m = 16; m >= 1; m >>= 1) acc += __shfl_xor(acc, m, 32);

    __shared__ float sm[8];
    int lane = threadIdx.x & 31;
    int wave = threadIdx.x >> 5;
    if (lane == 0) sm[wave] = acc;
    __syncthreads();

    if (wave == 0) {
        int nwaves = blockDim.x >> 5;
        float v = (lane < nwaves) ? sm[lane] : 0.0f;
        #pragma unroll
        for (int m = 4; m >= 1; m >>= 1) v += __shfl_xor(v, m, 32);
        if (lane == 0) partial[blockIdx.x] = v;
    }
}

// One wave (32 threads). Sums nPart partials; WMMA does the 64-way cross-lane
// reduction: D = ones(16x4) x B(4x16) + 0  =>  sum over row 0 of D == sum of
// every element of B regardless of the exact lane<->(k,n) mapping.
__global__ void quat_loss_final(const float* __restrict__ partial,
                                float* __restrict__ out,
                                int nPart, float invCount) {
    int lane = threadIdx.x;  // 0..31

    float a0 = 0.0f, a1 = 0.0f;
    int iters = (nPart + 63) >> 6;
    for (int k = 0; k < iters; ++k) {   // uniform trip count: EXEC full at WMMA
        int i = lane + 64 * k;
        if (i < nPart)      a0 += partial[i];
        if (i + 32 < nPart) a1 += partial[i + 32];
    }

    v2f a; a[0] = 1.0f; a[1] = 1.0f;     // A = ones(16x4)
    v2f b; b[0] = a0;   b[1] = a1;       // B = 4x16 holding all 64 lane accs
    v8f c = {};
    // D = A x B + 0  ->  v_wmma_f32_16x16x4_f32
    v8f d = __builtin_amdgcn_wmma_f32_16x16x4_f32(
        /*neg_a=*/false, a, /*neg_b=*/false, b,
        /*c_mod=*/(short)0, c, /*reuse_a=*/false, /*reuse_b=*/false);

    // Row 0 (d[0]) lanes 0..15 hold the 16 column sums; fold them.
    float s = d[0];
    s += __shfl_xor(s, 8, 32);
    s += __shfl_xor(s, 4, 32);
    s += __shfl_xor(s, 2, 32);
    s += __shfl_xor(s, 1, 32);

    if (lane == 0) out[0] = s * invCount;
}

extern "C" void kernel_launch(void* const* d_in, const int* in_sizes, int n_in,
                              void* d_out, int out_size, void* d_ws, size_t ws_size,
                              hipStream_t stream) {
    const float4* tq = (const float4*)d_in[0];  // true_quaternions  (B,4) f32
    const float4* pq = (const float4*)d_in[1];  // predicted_quaternion (B,4) f32
    // d_in[2] (batch_X) is unused by the reference math.

    int n = in_sizes[0] / 4;  // B

    int nblk = 1024;
    size_t needed = (size_t)nblk * sizeof(float);
    if (needed > ws_size) {
        nblk = (int)(ws_size / sizeof(float));
        if (nblk < 1) nblk = 1;
    }
    float* partial = (float*)d_ws;

    quat_loss_partial<<<nblk, 256, 0, stream>>>(tq, pq, partial, n);

    float invCount = 1.0f / (3.0f * (float)n);
    quat_loss_final<<<1, 32, 0, stream>>>(partial, (float*)d_out, nblk, invCount);
}